// CRF_88587995447877
// MI455X (gfx1250) — compile-verified
//
#include <hip/hip_runtime.h>
#include <hip/hip_bf16.h>
#include <cstddef>

// CRF forward-algorithm on gfx1250 (MI455X).
// Per-step log-semiring matvec -> real f32 GEMM via max-shift trick:
//   nxt_i = obs_i + m_b + log( sum_j exp(T_ij) * exp(prevs_j - m_b) )
// executed with V_WMMA_F32_16X16X4_F32 (K padded 66->68 => 17 WMMAs/N-tile,
// 5 N-tiles, one wave each, B-fragments resident in VGPRs).
// Emission tiles are software-pipelined: global loads for step t+1 are issued
// before step t's WMMA phase and published to LDS after the mid-step barrier.

typedef __attribute__((ext_vector_type(2))) float v2f;
typedef __attribute__((ext_vector_type(8))) float v8f;

#define TAGN   64
#define NEXT   66      // 64 tags + START + END
#define STARTI 64
#define ENDI   65
#define NEGV   (-10000.0f)
#define TLEN   1024
#define BTILE  16      // batches per block
#define PITCH  84      // LDS row pitch for prevs/vmat (no 64-bank walk)
#define KPAD   68      // K padded to multiple of 4
#define KCH    17      // 68/4 K-chunks
#define NTILES 5       // ceil(66/16) N-tiles
#define ETP    80      // E^T row pitch

__global__ __launch_bounds__(256) void crf_fwd_kernel(
    const float* __restrict__ logits,          // (B, T, 64)
    const unsigned char* __restrict__ maskg,   // (B, T) bool
    const int* __restrict__ tags,              // (B, T)
    const float* __restrict__ trans,           // (66, 66) row-major
    float* __restrict__ part_out)              // (B/16,) per-block sums
{
    __shared__ float  ET[KPAD * ETP];        // ET[j*ETP+i] = exp(trans[i][j]), 0-padded
    __shared__ float  prevs[BTILE * PITCH];  // log-domain state; padded cols stay NEG
    __shared__ float  vmat [BTILE * PITCH];  // exp(prevs - m)
    __shared__ float4 obs4 [BTILE * TAGN / 4]; // emission tile (current t)
    __shared__ float  mvec [BTILE];
    __shared__ float  real_s[BTILE];
    __shared__ float  diff_s[BTILE];
    __shared__ int    mask_s[BTILE];

    const float* obs_s = (const float*)obs4;

    const int tid    = threadIdx.x;
    const int lane   = tid & 31;
    const int wave   = tid >> 5;
    const int bb     = tid >> 4;   // batch within tile (16 lanes per batch)
    const int sub    = tid & 15;
    const int batch0 = blockIdx.x * BTILE;

    // my slice of the emission tile: 4 contiguous floats (16B coalesced)
    const int lb = tid >> 4;             // batch row of my float4
    const int li = (tid & 15) * 4;       // starting column

    // ---- one-time: exp(transitions)^T (padded) + prevs init ----
    for (int idx = tid; idx < KPAD * ETP; idx += 256) {
        int j = idx / ETP, i = idx % ETP;
        ET[idx] = (i < NEXT && j < NEXT) ? __expf(trans[i * NEXT + j]) : 0.0f;
    }
    for (int idx = tid; idx < BTILE * PITCH; idx += 256) prevs[idx] = NEGV;
    __syncthreads();
    if (tid < BTILE) prevs[tid * PITCH + STARTI] = 0.0f;

    // ---- resident B-fragments: wave w owns N-tile w ----
    v2f bfrag[KCH];
    if (wave < NTILES) {
        const int ncol = wave * 16 + (lane & 15);
        const int krow = (lane >> 4) * 2;   // lanes 0-15: K=0,1 ; 16-31: K=2,3
        #pragma unroll
        for (int k = 0; k < KCH; ++k) {
            int j = k * 4 + krow;
            bfrag[k].x = ET[j * ETP + ncol];
            bfrag[k].y = ET[(j + 1) * ETP + ncol];
        }
    }

    // ---- gold-path ("real") score: 16 lanes/batch, stride-16 over T ----
    {
        const size_t rowB = (size_t)(batch0 + bb) * TLEN;
        float acc = 0.0f;
        for (int ts = sub; ts < TLEN; ts += 16) {
            int mt  = maskg[rowB + ts];
            int tg  = tags [rowB + ts];
            int tgm = mt ? tg : ENDI;
            if (mt) acc += logits[(rowB + ts) * TAGN + tg];
            int ntag;
            if (ts == TLEN - 1) ntag = ENDI;
            else {
                int m1 = maskg[rowB + ts + 1];
                ntag = m1 ? tags[rowB + ts + 1] : ENDI;
            }
            if (mt) acc += trans[ntag * NEXT + tgm];
        }
        if (sub == 0) {
            int m0 = maskg[rowB];
            int t0 = m0 ? tags[rowB] : ENDI;
            acc += trans[t0 * NEXT + STARTI];
        }
        #pragma unroll
        for (int off = 8; off; off >>= 1) acc += __shfl_xor(acc, off, 32);
        if (sub == 0) real_s[bb] = acc;
    }
    __syncthreads();

    // ---- initial shift + exp buffer ----
    {
        float mx = -3.0e38f;
        const int i0 = sub * 5;
        #pragma unroll
        for (int q = 0; q < 5; ++q) mx = fmaxf(mx, prevs[bb * PITCH + i0 + q]);
        #pragma unroll
        for (int off = 8; off; off >>= 1) mx = fmaxf(mx, __shfl_xor(mx, off, 32));
        #pragma unroll
        for (int q = 0; q < 5; ++q)
            vmat[bb * PITCH + i0 + q] = __expf(prevs[bb * PITCH + i0 + q] - mx);
        if (sub == 0) mvec[bb] = mx;
    }

    // ---- prime the emission pipeline with t=0 ----
    {
        float4 p0 = *(const float4*)&logits[((size_t)(batch0 + lb) * TLEN + 0) * TAGN + li];
        int    m0 = (tid < BTILE) ? (int)maskg[(size_t)(batch0 + tid) * TLEN + 0] : 0;
        obs4[tid] = p0;
        if (tid < BTILE) mask_s[tid] = m0;
    }
    __syncthreads();

    // =============== sequential forward recurrence over T ===============
    for (int t = 0; t < TLEN; ++t) {
        // issue global loads for step t+1 NOW; consumed after the mid barrier,
        // so HBM latency overlaps the entire WMMA/log/max/exp phase.
        float4 pf = {0.f, 0.f, 0.f, 0.f};
        int    pfm = 0;
        const bool havenext = (t + 1 < TLEN);
        if (havenext) {
            pf = *(const float4*)&logits[((size_t)(batch0 + lb) * TLEN + (t + 1)) * TAGN + li];
            if (tid < BTILE) pfm = (int)maskg[(size_t)(batch0 + tid) * TLEN + (t + 1)];
        }

        if (wave < NTILES) {
            // D(16x16) = V(16x68) @ E^T(68x16): two independent WMMA chains
            v8f acc0 = {}, acc1 = {};
            const int mrow = lane & 15;
            const int krow = (lane >> 4) * 2;
            #pragma unroll
            for (int k = 0; k < KCH; ++k) {
                v2f a;
                a.x = vmat[mrow * PITCH + k * 4 + krow];
                a.y = vmat[mrow * PITCH + k * 4 + krow + 1];
                if (k & 1)
                    acc1 = __builtin_amdgcn_wmma_f32_16x16x4_f32(
                        false, a, false, bfrag[k], (short)0, acc1, false, false);
                else
                    acc0 = __builtin_amdgcn_wmma_f32_16x16x4_f32(
                        false, a, false, bfrag[k], (short)0, acc0, false, false);
            }
            // nxt[b,i] = obs + m_b + log(sum); masked rows keep prevs
            const int icol = wave * 16 + mrow;
            if (icol < NEXT) {
                const int bofs = (lane >> 4) * 8;   // lanes>=16 hold D rows M=8..15
                #pragma unroll
                for (int r = 0; r < 8; ++r) {
                    int b = bofs + r;
                    if (mask_s[b]) {
                        float o = (icol < TAGN) ? obs_s[b * TAGN + icol] : 0.0f;
                        float d = acc0[r] + acc1[r];
                        prevs[b * PITCH + icol] = mvec[b] + __logf(d) + o;
                    }
                }
            }
        }
        __syncthreads();

        // rebuild shift + exp buffer for next step
        {
            float mx = -3.0e38f;
            const int i0 = sub * 5;
            #pragma unroll
            for (int q = 0; q < 5; ++q) mx = fmaxf(mx, prevs[bb * PITCH + i0 + q]);
            #pragma unroll
            for (int off = 8; off; off >>= 1) mx = fmaxf(mx, __shfl_xor(mx, off, 32));
            #pragma unroll
            for (int q = 0; q < 5; ++q)
                vmat[bb * PITCH + i0 + q] = __expf(prevs[bb * PITCH + i0 + q] - mx);
            if (sub == 0) mvec[bb] = mx;
        }

        // publish the prefetched t+1 emission tile (waits on loadcnt here)
        if (havenext) {
            obs4[tid] = pf;
            if (tid < BTILE) mask_s[tid] = pfm;
        }
        __syncthreads();
    }

    // ---- norm = logsumexp_i(prevs + trans[END, i]); partial = norm - real ----
    {
        float mx = -3.0e38f;
        const int i0 = sub * 5;
        #pragma unroll
        for (int q = 0; q < 5; ++q) {
            int i = i0 + q;
            if (i < NEXT) mx = fmaxf(mx, prevs[bb * PITCH + i] + trans[ENDI * NEXT + i]);
        }
        #pragma unroll
        for (int off = 8; off; off >>= 1) mx = fmaxf(mx, __shfl_xor(mx, off, 32));
        float s = 0.0f;
        #pragma unroll
        for (int q = 0; q < 5; ++q) {
            int i = i0 + q;
            if (i < NEXT) s += __expf(prevs[bb * PITCH + i] + trans[ENDI * NEXT + i] - mx);
        }
        #pragma unroll
        for (int off = 8; off; off >>= 1) s += __shfl_xor(s, off, 32);
        if (sub == 0) diff_s[bb] = mx + __logf(s) - real_s[bb];
    }
    __syncthreads();
    if (tid == 0) {
        float tot = 0.0f;
        #pragma unroll
        for (int b = 0; b < BTILE; ++b) tot += diff_s[b];
        part_out[blockIdx.x] = tot;
    }
}

__global__ void crf_reduce_kernel(const float* __restrict__ part,
                                  float* __restrict__ out,
                                  int nparts, float invB)
{
    float v = (threadIdx.x < nparts) ? part[threadIdx.x] : 0.0f;
    #pragma unroll
    for (int off = 16; off; off >>= 1) v += __shfl_xor(v, off, 32);
    if (threadIdx.x == 0) out[0] = v * invB;
}

extern "C" void kernel_launch(void* const* d_in, const int* in_sizes, int n_in,
                              void* d_out, int out_size, void* d_ws, size_t ws_size,
                              hipStream_t stream)
{
    const float*         logits = (const float*)d_in[0];
    const unsigned char* maskg  = (const unsigned char*)d_in[1];
    const int*           tags   = (const int*)d_in[2];
    const float*         trans  = (const float*)d_in[3];

    const int B       = in_sizes[1] / TLEN;   // mask has B*T elements
    const int nblocks = B / BTILE;            // 32 for B=512
    float* parts = (float*)d_ws;

    crf_fwd_kernel<<<nblocks, 256, 0, stream>>>(logits, maskg, tags, trans, parts);
    crf_reduce_kernel<<<1, 32, 0, stream>>>(parts, (float*)d_out, nblocks, 1.0f / (float)B);
}